// PolicyNetwork_64527588655232
// MI455X (gfx1250) — compile-verified
//
#include <hip/hip_runtime.h>
#include <hip/hip_bf16.h>
#include <math.h>

typedef float v2f __attribute__((ext_vector_type(2)));
typedef float v8f __attribute__((ext_vector_type(8)));

// ---------------------------------------------------------------------------
// f32 WMMA: D(16x16) = A(16x4) * B(4x16) + C   (V_WMMA_F32_16X16X4_F32)
// Layouts (CDNA5 ISA 7.12.2, wave32):
//   A: lane L holds row (L&15); float2 = A[row][k0 + 2*(L>>4) + {0,1}].
//   B: lane L holds col (L&15); float2 = B[k0 + 2*(L>>4) + {0,1}][col].
//   C/D: vgpr v, lane L -> element (m = v + 8*(L>>4), n = L&15).
// ---------------------------------------------------------------------------
__device__ inline v8f wmma_f32_16x16x4(v2f a, v2f b, v8f c) {
    return __builtin_amdgcn_wmma_f32_16x16x4_f32(
        /*neg_a=*/false, a, /*neg_b=*/false, b,
        /*c_mod=*/(short)0, c, /*reuse_a=*/false, /*reuse_b=*/false);
}

__device__ inline void atomAddF(float* p, float v) {
    __hip_atomic_fetch_add(p, v, __ATOMIC_RELAXED, __HIP_MEMORY_SCOPE_AGENT);
}

// ---------------------------------------------------------------------------
// Stage 1: edge scatter for layer 1 (8 features) + degree.
// One thread per edge. x rows are 32B -> two float4 loads.
// ---------------------------------------------------------------------------
__global__ void sage_scatter1(const int* __restrict__ ei, const float* __restrict__ x,
                              float* __restrict__ agg1, float* __restrict__ deg, int E) {
    int e = blockIdx.x * blockDim.x + threadIdx.x;
    if (e >= E) return;
    int s = ei[e];
    int d = ei[E + e];
    const float4* xs = (const float4*)(x + (size_t)s * 8);
    float4 lo = xs[0];
    float4 hi = xs[1];
    float* a = agg1 + (size_t)d * 8;
    atomAddF(a + 0, lo.x); atomAddF(a + 1, lo.y);
    atomAddF(a + 2, lo.z); atomAddF(a + 3, lo.w);
    atomAddF(a + 4, hi.x); atomAddF(a + 5, hi.y);
    atomAddF(a + 6, hi.z); atomAddF(a + 7, hi.w);
    atomAddF(deg + d, 1.0f);
}

// ---------------------------------------------------------------------------
// Stage 2: layer-1 SAGEConv via WMMA. One wave = 16 nodes x 64 outputs.
// h1 = relu( (agg1/deg) @ W1_l + b1 + x @ W1_r )
// Branchless tail handling: clamp node index (always in-bounds), scale by
// a 0/1 row mask (folded into invd for the mean term) -> no EXEC diamonds
// in the k-loop, WMMAs stay in a full-EXEC straight line.
// ---------------------------------------------------------------------------
__global__ void sage_layer1(const float* __restrict__ x, const float* __restrict__ agg1,
                            const float* __restrict__ deg, const float* __restrict__ W1l,
                            const float* __restrict__ b1, const float* __restrict__ W1r,
                            float* __restrict__ h1, int N, int nTiles) {
    int wave = (int)((blockIdx.x * blockDim.x + threadIdx.x) >> 5);
    if (wave >= nTiles) return;                    // wave-uniform
    int lane  = threadIdx.x & 31;
    int col   = lane & 15;                         // A row index == B col index
    int khalf = lane >> 4;
    int rowNode  = wave * 16 + col;
    int nodeClmp = min(rowNode, N - 1);
    float rowScale = (rowNode < N) ? 1.0f : 0.0f;
    float invd = rowScale / fmaxf(deg[nodeClmp], 1.0f);

    const float* aggRow = agg1 + (size_t)nodeClmp * 8;
    const float* xRow   = x    + (size_t)nodeClmp * 8;

    v8f acc[4];
#pragma unroll
    for (int nt = 0; nt < 4; ++nt) {
        float bv = b1[nt * 16 + col];
#pragma unroll
        for (int v = 0; v < 8; ++v) acc[nt][v] = bv;
    }

    // mean-aggregate term: (agg1 * invd) @ W1_l   (K = 8 -> two k-steps)
#pragma unroll
    for (int k0 = 0; k0 < 8; k0 += 4) {
        int ka = k0 + 2 * khalf;
        v2f a;
        a.x = aggRow[ka + 0] * invd;
        a.y = aggRow[ka + 1] * invd;
#pragma unroll
        for (int nt = 0; nt < 4; ++nt) {
            v2f b;
            b.x = W1l[(ka + 0) * 64 + nt * 16 + col];
            b.y = W1l[(ka + 1) * 64 + nt * 16 + col];
            acc[nt] = wmma_f32_16x16x4(a, b, acc[nt]);
        }
    }
    // root term: x @ W1_r
#pragma unroll
    for (int k0 = 0; k0 < 8; k0 += 4) {
        int ka = k0 + 2 * khalf;
        v2f a;
        a.x = xRow[ka + 0] * rowScale;
        a.y = xRow[ka + 1] * rowScale;
#pragma unroll
        for (int nt = 0; nt < 4; ++nt) {
            v2f b;
            b.x = W1r[(ka + 0) * 64 + nt * 16 + col];
            b.y = W1r[(ka + 1) * 64 + nt * 16 + col];
            acc[nt] = wmma_f32_16x16x4(a, b, acc[nt]);
        }
    }
    // relu + store h1 (C/D layout: m = v + 8*khalf)
#pragma unroll
    for (int nt = 0; nt < 4; ++nt) {
#pragma unroll
        for (int v = 0; v < 8; ++v) {
            int m = wave * 16 + v + 8 * khalf;
            if (m < N) h1[(size_t)m * 64 + nt * 16 + col] = fmaxf(acc[nt][v], 0.0f);
        }
    }
}

// ---------------------------------------------------------------------------
// Stage 3: edge scatter for layer 2 (64 features). 16 threads per edge,
// each thread moves a float4 (h1 and agg2 are L2-resident on MI455X).
// ---------------------------------------------------------------------------
__global__ void sage_scatter2(const int* __restrict__ ei, const float* __restrict__ h1,
                              float* __restrict__ agg2, int E) {
    int g = blockIdx.x * blockDim.x + threadIdx.x;
    int e = g >> 4;
    if (e >= E) return;
    int part = g & 15;
    int s = ei[e];
    int d = ei[E + e];
    float4 v = *(const float4*)(h1 + (size_t)s * 64 + part * 4);
    float* a = agg2 + (size_t)d * 64 + part * 4;
    atomAddF(a + 0, v.x); atomAddF(a + 1, v.y);
    atomAddF(a + 2, v.z); atomAddF(a + 3, v.w);
}

// ---------------------------------------------------------------------------
// Stage 4: layer-2 SAGEConv via WMMA (K = 64 per matmul), fused with the
// global mean-pool reduction. h2 is never materialized. Branchless k-loop.
// ---------------------------------------------------------------------------
__global__ void sage_layer2_pool(const float* __restrict__ h1, const float* __restrict__ agg2,
                                 const float* __restrict__ deg, const float* __restrict__ W2l,
                                 const float* __restrict__ b2, const float* __restrict__ W2r,
                                 float* __restrict__ pooled, int N, int nTiles) {
    int wave = (int)((blockIdx.x * blockDim.x + threadIdx.x) >> 5);
    if (wave >= nTiles) return;                    // wave-uniform
    int lane  = threadIdx.x & 31;
    int col   = lane & 15;
    int khalf = lane >> 4;
    int rowNode  = wave * 16 + col;
    int nodeClmp = min(rowNode, N - 1);
    float rowScale = (rowNode < N) ? 1.0f : 0.0f;
    float invd = rowScale / fmaxf(deg[nodeClmp], 1.0f);

    const float* aggRow = agg2 + (size_t)nodeClmp * 64;
    const float* h1Row  = h1   + (size_t)nodeClmp * 64;

    v8f acc[4];
#pragma unroll
    for (int nt = 0; nt < 4; ++nt) {
        float bv = b2[nt * 16 + col];
#pragma unroll
        for (int v = 0; v < 8; ++v) acc[nt][v] = bv;
    }

    // (agg2/deg) @ W2_l : 16 k-steps
    for (int k0 = 0; k0 < 64; k0 += 4) {
        int ka = k0 + 2 * khalf;
        v2f a;
        a.x = aggRow[ka + 0] * invd;
        a.y = aggRow[ka + 1] * invd;
#pragma unroll
        for (int nt = 0; nt < 4; ++nt) {
            v2f b;
            b.x = W2l[(ka + 0) * 64 + nt * 16 + col];
            b.y = W2l[(ka + 1) * 64 + nt * 16 + col];
            acc[nt] = wmma_f32_16x16x4(a, b, acc[nt]);
        }
    }
    // h1 @ W2_r : 16 k-steps
    for (int k0 = 0; k0 < 64; k0 += 4) {
        int ka = k0 + 2 * khalf;
        v2f a;
        a.x = h1Row[ka + 0] * rowScale;
        a.y = h1Row[ka + 1] * rowScale;
#pragma unroll
        for (int nt = 0; nt < 4; ++nt) {
            v2f b;
            b.x = W2r[(ka + 0) * 64 + nt * 16 + col];
            b.y = W2r[(ka + 1) * 64 + nt * 16 + col];
            acc[nt] = wmma_f32_16x16x4(a, b, acc[nt]);
        }
    }
    // relu + column-sum this tile's contribution to the global mean pool
#pragma unroll
    for (int nt = 0; nt < 4; ++nt) {
        float csum = 0.0f;
#pragma unroll
        for (int v = 0; v < 8; ++v) {
            int m = wave * 16 + v + 8 * khalf;
            float val = fmaxf(acc[nt][v], 0.0f);
            csum += (m < N) ? val : 0.0f;
        }
        atomAddF(&pooled[nt * 16 + col], csum);
    }
}

// ---------------------------------------------------------------------------
// Stage 5: MLP head + softmax on the pooled (1,64) vector. One small block.
// ---------------------------------------------------------------------------
__global__ void sage_head(const float* __restrict__ pooled,
                          const float* __restrict__ fc1W, const float* __restrict__ fc1b,
                          const float* __restrict__ fc2W, const float* __restrict__ fc2b,
                          float* __restrict__ out, int N) {
    __shared__ float g[64];
    __shared__ float a1[64];
    __shared__ float logits[10];
    int t = threadIdx.x;
    if (t < 64) g[t] = pooled[t] * (1.0f / (float)N);
    __syncthreads();
    if (t < 64) {
        float s = fc1b[t];
        for (int k = 0; k < 64; ++k) s += g[k] * fc1W[k * 64 + t];
        a1[t] = fmaxf(s, 0.0f);
    }
    __syncthreads();
    if (t < 10) {
        float s = fc2b[t];
        for (int k = 0; k < 64; ++k) s += a1[k] * fc2W[k * 10 + t];
        logits[t] = s;
    }
    __syncthreads();
    if (t == 0) {
        float mx = logits[0];
        for (int i = 1; i < 10; ++i) mx = fmaxf(mx, logits[i]);
        float se = 0.0f;
        float ex[10];
        for (int i = 0; i < 10; ++i) { ex[i] = __expf(logits[i] - mx); se += ex[i]; }
        float inv = 1.0f / se;
        for (int i = 0; i < 10; ++i) out[i] = ex[i] * inv;
    }
}

// ---------------------------------------------------------------------------
extern "C" void kernel_launch(void* const* d_in, const int* in_sizes, int n_in,
                              void* d_out, int out_size, void* d_ws, size_t ws_size,
                              hipStream_t stream) {
    const float* x    = (const float*)d_in[0];
    const int*   ei   = (const int*)  d_in[1];
    const float* W1l  = (const float*)d_in[2];
    const float* b1   = (const float*)d_in[3];
    const float* W1r  = (const float*)d_in[4];
    const float* W2l  = (const float*)d_in[5];
    const float* b2   = (const float*)d_in[6];
    const float* W2r  = (const float*)d_in[7];
    const float* fc1W = (const float*)d_in[8];
    const float* fc1b = (const float*)d_in[9];
    const float* fc2W = (const float*)d_in[10];
    const float* fc2b = (const float*)d_in[11];

    int N = in_sizes[0] / 8;
    int E = in_sizes[1] / 2;
    int nTiles = (N + 15) / 16;
    size_t Np = (size_t)nTiles * 16;

    float* ws     = (float*)d_ws;
    float* deg    = ws;                       // Np
    float* agg1   = deg  + Np;                // Np*8
    float* h1     = agg1 + Np * 8;            // Np*64
    float* agg2   = h1   + Np * 64;           // Np*64
    float* pooled = agg2 + Np * 64;           // 64
    size_t totalFloats = Np * (1 + 8 + 64 + 64) + 64;

    hipMemsetAsync(d_ws, 0, totalFloats * sizeof(float), stream);

    sage_scatter1<<<(E + 255) / 256, 256, 0, stream>>>(ei, x, agg1, deg, E);

    sage_layer1<<<(nTiles + 7) / 8, 256, 0, stream>>>(x, agg1, deg, W1l, b1, W1r, h1, N, nTiles);

    long long work2 = (long long)E * 16;
    sage_scatter2<<<(int)((work2 + 255) / 256), 256, 0, stream>>>(ei, h1, agg2, E);

    sage_layer2_pool<<<(nTiles + 7) / 8, 256, 0, stream>>>(h1, agg2, deg, W2l, b2, W2r,
                                                           pooled, N, nTiles);

    sage_head<<<1, 64, 0, stream>>>(pooled, fc1W, fc1b, fc2W, fc2b, (float*)d_out, N);
}